// DynResNet_53807350284653
// MI455X (gfx1250) — compile-verified
//
#include <hip/hip_runtime.h>

// ---------------------------------------------------------------------------
// DynResNet on MI455X (gfx1250): fp32 WMMA GEMM chain (2x2 register blocking,
// TDM-fed W tiles, float4 staging) + per-sample low-rank Cayley step with the
// 28x28 pinv removed via the push-through/Woodbury identity.
// ---------------------------------------------------------------------------

#define NSAMP 2048
#define DIMN  28
#define KRr   10
#define NFEAT 784      // 28*28
#define NLAY  32
#define KCLSn 10
#define HSTEP 0.001f
#define TSTEPS 33      // 1 + NLAY

typedef __attribute__((ext_vector_type(2))) float v2f;
typedef __attribute__((ext_vector_type(8))) float v8f;
typedef __attribute__((ext_vector_type(4))) unsigned int v4u;
typedef __attribute__((ext_vector_type(8))) int v8i;
typedef __attribute__((ext_vector_type(4))) int v4i;

#if defined(__has_builtin)
# if __has_builtin(__builtin_amdgcn_tensor_load_to_lds)
#  define HAVE_TDM 1
# endif
#endif
#ifndef HAVE_TDM
# define HAVE_TDM 0
#endif

#if HAVE_TDM
// Issue one TDM 2-D tile load: tile (tile0 x tile1) dwords from a row-major
// f32 tensor with row stride `stride0`, into LDS at byte offset lds_off with
// padded rows (32 dwords data + 4 dwords pad -> stride 36 dwords).
// tensor_dim* are the remaining extents measured from the tile origin, so the
// TDM zero-fills out-of-bounds rows/cols for us (ISA 8.2: OOB reads -> 0).
__device__ inline void tdm_load_tile_f32(unsigned int lds_off, const float* gptr,
                                         int tdim0, int tdim1, int tile0, int tile1,
                                         int stride0)
{
    unsigned long long ga = (unsigned long long)(uintptr_t)gptr;
    v4u g0;
    g0.x = 1u;                                            // count=1 (valid user D#)
    g0.y = lds_off;                                       // lds_addr (bytes)
    g0.z = (unsigned int)(ga & 0xFFFFFFFFu);              // global_addr[31:0]
    g0.w = (unsigned int)((ga >> 32) & 0x01FFFFFFu)       // global_addr[56:32]
         | (2u << 30);                                    // type = 2 ("image")
    v8i g1;
    g1[0] = (2 << 16)        // data_size = 4B
          | (1 << 20)        // pad_enable
          | (4 << 22)        // pad_interval: 32 dwords
          | (3 << 25);       // pad_amount: 4 dwords  -> LDS row stride 36 dwords
    g1[1] = (tdim0 & 0xFFFF) << 16;                        // tensor_dim0[15:0]
    g1[2] = ((tdim0 >> 16) & 0xFFFF) | ((tdim1 & 0xFFFF) << 16);
    g1[3] = ((tdim1 >> 16) & 0xFFFF) | ((tile0 & 0xFFFF) << 16);
    g1[4] = (tile1 & 0xFFFF);                              // tile_dim1, tile_dim2=0
    g1[5] = stride0;                                       // tensor_dim0_stride[31:0]
    g1[6] = 0;                                             // stride0 hi / stride1 lo
    g1[7] = 0;
    v4i z4 = {0, 0, 0, 0};
# if __clang_major__ >= 23
    v8i z8 = {0, 0, 0, 0, 0, 0, 0, 0};
    __builtin_amdgcn_tensor_load_to_lds(g0, g1, z4, z4, z8, 0);
# else
    __builtin_amdgcn_tensor_load_to_lds(g0, g1, z4, z4, 0);
# endif
}
#endif

// ---------------------------------------------------------------------------
// GEMM: Y[m][n] = act( sum_k X[m][k] * W[n][k] + b[n] )    (Y = X @ W^T)
// M=2048, N=K=784.  Block: 256 thr = 8 waves (4 M x 2 N); each wave owns a
// 32x32 macro-tile = 2x2 WMMA tiles (4 v8f accumulators, 4 wmma per k-step).
// BM=128, BN=64, BK=32.  LDS rows padded to 36 dwords: conflict-free fragment
// gathers (36 = 4 mod 64) and 16B-aligned float4 staging stores.
// ---------------------------------------------------------------------------
__global__ __launch_bounds__(256)
void dyn_gemm_wmma(const float* __restrict__ Xm, const float* __restrict__ Wm,
                   const float* __restrict__ bias, float* __restrict__ Ym,
                   int fuseBiasRelu)
{
    __shared__ __align__(16) float xs[128][36];
    __shared__ __align__(16) float wt[64][36];

    const int tid  = threadIdx.x;
    const int lane = tid & 31;
    const int wid  = tid >> 5;
    const int lm   = lane & 15;     // lane % 16
    const int half = lane >> 4;     // A/B: K {0,1} vs {2,3}; C: M 0..7 vs 8..15
    const int tmL  = wid >> 1;      // 0..3 -> M offset tmL*32
    const int tnL  = wid & 1;       // 0..1 -> N offset tnL*32

    const int xrow0 = blockIdx.y * 128;
    const int wrow0 = blockIdx.x * 64;

    v8f acc00 = {}, acc01 = {}, acc10 = {}, acc11 = {};

    for (int kb = 0; kb < NFEAT; kb += 32) {
#if HAVE_TDM
        // One wave launches the DMA for the W tile; overlaps with X staging.
        if (wid == 0) {
            tdm_load_tile_f32((unsigned int)(uintptr_t)&wt[0][0],
                              Wm + (size_t)wrow0 * NFEAT + kb,
                              NFEAT - kb, NFEAT - wrow0, 32, 64, NFEAT);
        }
#endif
        // Stage X tile 128x32 with float4 loads (whole-vector OOB guard:
        // 784 % 4 == 0, so each float4 is fully valid or fully invalid).
#pragma unroll
        for (int t = 0; t < 4; ++t) {
            const int idx = tid + t * 256;
            const int r = idx >> 3, c4 = (idx & 7) * 4;
            const int k = kb + c4;
            float4 val = make_float4(0.f, 0.f, 0.f, 0.f);
            if (k < NFEAT)
                val = *(const float4*)&Xm[(size_t)(xrow0 + r) * NFEAT + k];
            *(float4*)&xs[r][c4] = val;
        }
#if !HAVE_TDM
        // Fallback: stage W tile 64x32 with guarded float4 loads.
#pragma unroll
        for (int t = 0; t < 2; ++t) {
            const int idx = tid + t * 256;
            const int r = idx >> 3, c4 = (idx & 7) * 4;
            const int k = kb + c4, wr = wrow0 + r;
            float4 val = make_float4(0.f, 0.f, 0.f, 0.f);
            if (wr < NFEAT && k < NFEAT)
                val = *(const float4*)&Wm[(size_t)wr * NFEAT + k];
            *(float4*)&wt[r][c4] = val;
        }
#endif
#if HAVE_TDM
        if (wid == 0) __builtin_amdgcn_s_wait_tensorcnt(0);
#endif
        __syncthreads();

#pragma unroll
        for (int kk = 0; kk < 8; ++kk) {
            const int kc = kk * 4 + 2 * half;
            v2f a0, a1, b0, b1;
            a0.x = xs[tmL * 32 + lm][kc];        a0.y = xs[tmL * 32 + lm][kc + 1];
            a1.x = xs[tmL * 32 + 16 + lm][kc];   a1.y = xs[tmL * 32 + 16 + lm][kc + 1];
            b0.x = wt[tnL * 32 + lm][kc];        b0.y = wt[tnL * 32 + lm][kc + 1];
            b1.x = wt[tnL * 32 + 16 + lm][kc];   b1.y = wt[tnL * 32 + 16 + lm][kc + 1];
            acc00 = __builtin_amdgcn_wmma_f32_16x16x4_f32(false, a0, false, b0, (short)0, acc00, false, false);
            acc01 = __builtin_amdgcn_wmma_f32_16x16x4_f32(false, a0, false, b1, (short)0, acc01, false, false);
            acc10 = __builtin_amdgcn_wmma_f32_16x16x4_f32(false, a1, false, b0, (short)0, acc10, false, false);
            acc11 = __builtin_amdgcn_wmma_f32_16x16x4_f32(false, a1, false, b1, (short)0, acc11, false, false);
        }
        __syncthreads();
    }

    // C/D layout: VGPR i -> M = i + 8*half, N = lane%16
    const int rB = xrow0 + tmL * 32;
    const int cB = wrow0 + tnL * 32;
#pragma unroll
    for (int mi = 0; mi < 2; ++mi) {
#pragma unroll
        for (int ni = 0; ni < 2; ++ni) {
            const v8f acc = (mi == 0) ? (ni == 0 ? acc00 : acc01)
                                      : (ni == 0 ? acc10 : acc11);
            const int col = cB + ni * 16 + lm;
            if (col < NFEAT) {
                const float bv = fuseBiasRelu ? bias[col] : 0.f;
#pragma unroll
                for (int i = 0; i < 8; ++i) {
                    const int row = rB + mi * 16 + i + 8 * half;
                    float v = acc[i] + bv;
                    if (fuseBiasRelu) v = fmaxf(v, 0.f);
                    Ym[(size_t)row * NFEAT + col] = v;
                }
            }
        }
    }
}

// ---------------------------------------------------------------------------
// Init: unpack u (28x10), s (10x10), v (28x10, = vh^T) from X; x = u @ s @ vh
// ---------------------------------------------------------------------------
__global__ __launch_bounds__(128)
void dyn_init(const float* __restrict__ Xin, float* __restrict__ uB,
              float* __restrict__ sB, float* __restrict__ vB,
              float* __restrict__ xB)
{
    __shared__ float su[280], ss[100], sv[280], sy[280];
    const int n = blockIdx.x, tid = threadIdx.x;
    const float* base = Xin + (size_t)n * 3 * NFEAT;

    for (int idx = tid; idx < 280; idx += 128) su[idx] = base[idx];
    for (int idx = tid; idx < 100; idx += 128) {
        int a = idx / 10, b = idx % 10;                  // top 10x10 of (28,10)
        ss[idx] = base[NFEAT + a * 10 + b];
    }
    for (int idx = tid; idx < 280; idx += 128) {
        int i = idx / 10, j = idx % 10;                  // v[i][j] = vh[j][i]
        sv[idx] = base[2 * NFEAT + j * DIMN + i];
    }
    __syncthreads();

    for (int idx = tid; idx < 280; idx += 128) {         // y = u @ s
        int i = idx / 10, j = idx % 10;
        float acc = 0.f;
        for (int a = 0; a < 10; ++a) acc += su[i * 10 + a] * ss[a * 10 + j];
        sy[idx] = acc;
    }
    __syncthreads();

    for (int idx = tid; idx < NFEAT; idx += 128) {       // x = y @ v^T
        int i = idx / 28, j = idx % 28;
        float acc = 0.f;
        for (int a = 0; a < 10; ++a) acc += sy[i * 10 + a] * sv[j * 10 + a];
        xB[(size_t)n * NFEAT + idx] = acc;
    }
    float* up = uB + (size_t)n * 280;
    float* vp = vB + (size_t)n * 280;
    float* sp = sB + (size_t)n * 100;
    for (int idx = tid; idx < 280; idx += 128) { up[idx] = su[idx]; vp[idx] = sv[idx]; }
    for (int idx = tid; idx < 100; idx += 128) sp[idx] = ss[idx];
}

// ---------------------------------------------------------------------------
// Low-rank dynamics step. Cayley with DT == -C reduced via Woodbury:
//   u_new = u - H^2 * Fu @ ((u^T Fu) @ (u^T u))      (+O(H^4), < fp32 eps)
// ---------------------------------------------------------------------------
__global__ __launch_bounds__(128)
void dyn_step(const float* __restrict__ dY, float* __restrict__ uB,
              float* __restrict__ sB, float* __restrict__ vB,
              float* __restrict__ xB, float* __restrict__ outT, int tslot)
{
    __shared__ float sdy[784];
    __shared__ float su[280], sv[280], ss[100];
    __shared__ float t1[280], t2[280], fu[280], fv[280];
    __shared__ float g1[100], g2[100], g3[100], g4[100], g5[100];
    __shared__ float g6[100], g7[100], g8[100], g9[100];
    __shared__ float dinv[16];

    const int n = blockIdx.x, tid = threadIdx.x;
    float* up = uB + (size_t)n * 280;
    float* vp = vB + (size_t)n * 280;
    float* sp = sB + (size_t)n * 100;
    const float* dyp = dY + (size_t)n * NFEAT;

    for (int idx = tid; idx < 784; idx += 128) sdy[idx] = dyp[idx];
    for (int idx = tid; idx < 280; idx += 128) { su[idx] = up[idx]; sv[idx] = vp[idx]; }
    for (int idx = tid; idx < 100; idx += 128) ss[idx] = sp[idx];
    __syncthreads();

    // t1 = dY @ v ; t2 = dY^T @ u
    for (int idx = tid; idx < 280; idx += 128) {
        int i = idx / 10, j = idx % 10;
        float a1 = 0.f, a2 = 0.f;
        for (int k = 0; k < 28; ++k) {
            a1 += sdy[i * 28 + k] * sv[k * 10 + j];
            a2 += sdy[k * 28 + i] * su[k * 10 + j];
        }
        t1[idx] = a1; t2[idx] = a2;
    }
    if (tid < 10) dinv[tid] = 1.0f / ss[tid * 10 + tid];
    __syncthreads();

    // dS = u^T @ t1 (pre-scale)
    for (int idx = tid; idx < 100; idx += 128) {
        int a = idx / 10, b = idx % 10;
        float acc = 0.f;
        for (int k = 0; k < 28; ++k) acc += su[k * 10 + a] * t1[k * 10 + b];
        g1[idx] = acc;
    }
    __syncthreads();

    // dU = t1 * s_inv, dV = t2 * s_inv  (diagonal, column scale)
    for (int idx = tid; idx < 280; idx += 128) {
        float dv = dinv[idx % 10];
        t1[idx] *= dv; t2[idx] *= dv;
    }
    __syncthreads();

    // Gu = u^T dU ; Gv = v^T dV
    for (int idx = tid; idx < 100; idx += 128) {
        int a = idx / 10, b = idx % 10;
        float a1 = 0.f, a2 = 0.f;
        for (int k = 0; k < 28; ++k) {
            a1 += su[k * 10 + a] * t1[k * 10 + b];
            a2 += sv[k * 10 + a] * t2[k * 10 + b];
        }
        g2[idx] = a1; g3[idx] = a2;
    }
    __syncthreads();

    // Fu = dU - u Gu ; Fv = dV - v Gv
    for (int idx = tid; idx < 280; idx += 128) {
        int i = idx / 10, j = idx % 10;
        float a1 = 0.f, a2 = 0.f;
        for (int a = 0; a < 10; ++a) {
            a1 += su[i * 10 + a] * g2[a * 10 + j];
            a2 += sv[i * 10 + a] * g3[a * 10 + j];
        }
        fu[idx] = t1[idx] - a1; fv[idx] = t2[idx] - a2;
    }
    __syncthreads();

    // Gfu = u^T Fu, Guu = u^T u, Gfv = v^T Fv, Gvv = v^T v
    for (int idx = tid; idx < 100; idx += 128) {
        int a = idx / 10, b = idx % 10;
        float x1 = 0.f, x2 = 0.f, x3 = 0.f, x4 = 0.f;
        for (int k = 0; k < 28; ++k) {
            x1 += su[k * 10 + a] * fu[k * 10 + b];
            x2 += su[k * 10 + a] * su[k * 10 + b];
            x3 += sv[k * 10 + a] * fv[k * 10 + b];
            x4 += sv[k * 10 + a] * sv[k * 10 + b];
        }
        g4[idx] = x1; g5[idx] = x2; g6[idx] = x3; g7[idx] = x4;
    }
    __syncthreads();

    // Pu = Gfu @ Guu ; Pv = Gfv @ Gvv
    for (int idx = tid; idx < 100; idx += 128) {
        int a = idx / 10, b = idx % 10;
        float x1 = 0.f, x2 = 0.f;
        for (int k = 0; k < 10; ++k) {
            x1 += g4[a * 10 + k] * g5[k * 10 + b];
            x2 += g6[a * 10 + k] * g7[k * 10 + b];
        }
        g8[idx] = x1; g9[idx] = x2;
    }
    __syncthreads();

    // u -= H^2 Fu Pu ; v -= H^2 Fv Pv ; s += H dS
    const float h2 = HSTEP * HSTEP;
    for (int idx = tid; idx < 280; idx += 128) {
        int i = idx / 10, j = idx % 10;
        float x1 = 0.f, x2 = 0.f;
        for (int a = 0; a < 10; ++a) {
            x1 += fu[i * 10 + a] * g8[a * 10 + j];
            x2 += fv[i * 10 + a] * g9[a * 10 + j];
        }
        su[idx] -= h2 * x1; sv[idx] -= h2 * x2;
    }
    for (int idx = tid; idx < 100; idx += 128) ss[idx] += HSTEP * g1[idx];
    __syncthreads();

    // y = u_new @ s_new
    for (int idx = tid; idx < 280; idx += 128) {
        int i = idx / 10, j = idx % 10;
        float acc = 0.f;
        for (int a = 0; a < 10; ++a) acc += su[i * 10 + a] * ss[a * 10 + j];
        t1[idx] = acc;
    }
    __syncthreads();

    // x = y @ v_new^T ; scatter into X_transformed (stride TSTEPS)
    for (int idx = tid; idx < 784; idx += 128) {
        int i = idx / 28, j = idx % 28;
        float acc = 0.f;
        for (int a = 0; a < 10; ++a) acc += t1[i * 10 + a] * sv[j * 10 + a];
        xB[(size_t)n * NFEAT + idx] = acc;
        outT[(size_t)n * NFEAT * TSTEPS + (size_t)idx * TSTEPS + tslot] = acc;
    }
    for (int idx = tid; idx < 280; idx += 128) { up[idx] = su[idx]; vp[idx] = sv[idx]; }
    for (int idx = tid; idx < 100; idx += 128) sp[idx] = ss[idx];
}

// ---------------------------------------------------------------------------
// Classifier + softmax: one wave32 per sample, shuffle-reduce the 10 dots.
// ---------------------------------------------------------------------------
__global__ __launch_bounds__(32)
void dyn_cls(const float* __restrict__ xB, const float* __restrict__ Wc,
             const float* __restrict__ bc, float* __restrict__ pred,
             float* __restrict__ cls)
{
    const int n = blockIdx.x, lane = threadIdx.x;
    const float* xp = xB + (size_t)n * NFEAT;
    float v[10];
#pragma unroll
    for (int c = 0; c < 10; ++c) {
        float acc = 0.f;
        for (int k = lane; k < NFEAT; k += 32) acc += xp[k] * Wc[(size_t)c * NFEAT + k];
#pragma unroll
        for (int off = 16; off; off >>= 1) acc += __shfl_xor(acc, off, 32);
        v[c] = acc + bc[c];
    }
    float m = v[0];
#pragma unroll
    for (int c = 1; c < 10; ++c) m = fmaxf(m, v[c]);
    float e[10], sum = 0.f;
#pragma unroll
    for (int c = 0; c < 10; ++c) { e[c] = __expf(v[c] - m); sum += e[c]; }
    const float inv = 1.0f / sum;
    if (lane == 0) {
#pragma unroll
        for (int c = 0; c < 10; ++c) {
            cls[(size_t)n * 10 + c]  = v[c];
            pred[(size_t)n * 10 + c] = e[c] * inv;
        }
    }
}

// ---------------------------------------------------------------------------
extern "C" void kernel_launch(void* const* d_in, const int* in_sizes, int n_in,
                              void* d_out, int out_size, void* d_ws, size_t ws_size,
                              hipStream_t stream)
{
    (void)in_sizes; (void)n_in; (void)out_size; (void)ws_size;
    const float* X  = (const float*)d_in[0];
    const float* W0 = (const float*)d_in[1];
    const float* Ws = (const float*)d_in[2];
    const float* bs = (const float*)d_in[3];
    const float* Wc = (const float*)d_in[4];
    const float* bc = (const float*)d_in[5];

    float* out   = (float*)d_out;
    float* pred  = out;                                    // (N, 10)
    float* cls   = out + (size_t)NSAMP * KCLSn;            // (N, 10)
    float* trans = out + (size_t)2 * NSAMP * KCLSn;        // (N, 784, 33)

    float* wsf = (float*)d_ws;
    float* xB  = wsf;                                      // N*784
    float* dyB = xB  + (size_t)NSAMP * NFEAT;              // N*784
    float* uB  = dyB + (size_t)NSAMP * NFEAT;              // N*280
    float* vB  = uB  + (size_t)NSAMP * 280;                // N*280
    float* sB  = vB  + (size_t)NSAMP * 280;                // N*100

    const dim3 gGemm((NFEAT + 63) / 64, NSAMP / 128);      // (13, 16)

    dyn_init<<<NSAMP, 128, 0, stream>>>(X, uB, sB, vB, xB);
    dyn_gemm_wmma<<<gGemm, 256, 0, stream>>>(xB, W0, nullptr, dyB, 0);
    dyn_step<<<NSAMP, 128, 0, stream>>>(dyB, uB, sB, vB, xB, trans, 0);

    for (int l = 0; l < NLAY; ++l) {
        dyn_gemm_wmma<<<gGemm, 256, 0, stream>>>(
            xB, Ws + (size_t)l * NFEAT * NFEAT, bs + (size_t)l * NFEAT, dyB, 1);
        dyn_step<<<NSAMP, 128, 0, stream>>>(dyB, uB, sB, vB, xB, trans, l + 1);
    }

    dyn_cls<<<NSAMP, 32, 0, stream>>>(xB, Wc, bc, pred, cls);
}